// FEASAI_2619930051038
// MI455X (gfx1250) — compile-verified
//
#include <hip/hip_runtime.h>
#include <cstdint>
#include <cstddef>

// Problem constants (reference shapes: bs=4, ts=64, T27=27, H=W=256, f32).
constexpr int kW   = 256;
constexpr int kH   = 256;
constexpr int kTS  = 64;
constexpr int kT27 = 27;
#define EPSF 1e-3f

// ---- CDNA5 async global->LDS path (ASYNCcnt), guarded so we always compile ----
#if defined(__HIP_DEVICE_COMPILE__) && __has_builtin(__builtin_amdgcn_global_load_async_to_lds_b32)
#define FEASAI_ASYNC 1
#endif

#ifdef FEASAI_ASYNC
typedef __attribute__((address_space(1))) int* feasai_gptr;
typedef __attribute__((address_space(3))) int* feasai_lptr;
#endif

__device__ __forceinline__ void feasai_async_cp(const float* gsrc, float* ldst) {
#ifdef FEASAI_ASYNC
  // per-lane 4-byte async copy from global address gsrc to LDS address ldst, tracked by ASYNCcnt
  __builtin_amdgcn_global_load_async_to_lds_b32(
      (feasai_gptr)(void*)gsrc,
      (feasai_lptr)(void*)ldst,
      /*offset=*/0, /*cpol=*/0);
#else
  *ldst = *gsrc;  // synchronous fallback (no async builtin on this toolchain)
#endif
}

__device__ __forceinline__ void feasai_wait_async_le1() {
#ifdef FEASAI_ASYNC
#if __has_builtin(__builtin_amdgcn_s_wait_asynccnt)
  __builtin_amdgcn_s_wait_asynccnt(1);
#else
  asm volatile("s_wait_asynccnt 1" ::: "memory");
#endif
#endif
}

__device__ __forceinline__ void feasai_wait_async_le0() {
#ifdef FEASAI_ASYNC
#if __has_builtin(__builtin_amdgcn_s_wait_asynccnt)
  __builtin_amdgcn_s_wait_asynccnt(0);
#else
  asm volatile("s_wait_asynccnt 0" ::: "memory");
#endif
#endif
}

// reference: x0=floor(xp); w=xp-x0; x0i=clip(int(x0),0,W-1); x1i=clip(x0i+1,0,W-1)
__device__ __forceinline__ float feasai_bilin(const float* row, float xp) {
  float x0f = floorf(xp);
  float w   = xp - x0f;
  int x0i = (int)x0f;
  x0i = x0i < 0 ? 0 : (x0i > kW - 1 ? kW - 1 : x0i);
  int x1i = x0i + 1;
  x1i = x1i > kW - 1 ? kW - 1 : x1i;
  return (1.0f - w) * row[x0i] + w * row[x1i];
}

__global__ __launch_bounds__(kW) void feasai_refocus_kernel(
    const float* __restrict__ voxelgrid,  // [bs,64,H,W]
    const float* __restrict__ time,       // [bs,64]
    const float* __restrict__ occ_aps,    // [bs,27,H,W]
    const float* __restrict__ occ_t,      // [bs,27]
    const float* __restrict__ gt_t,       // [bs]
    const float* __restrict__ fx,         // [bs,3,3]
    const float* __restrict__ v,          // [bs]
    const float* __restrict__ depth_gt,   // [bs,27,H,W]
    const float* __restrict__ flow_27,    // [bs,27,H,W]
    float* __restrict__ out)              // [bs,6,H,W]
{
  const int y = blockIdx.x;   // row
  const int b = blockIdx.y;   // batch
  const int x = threadIdx.x;  // pixel

  __shared__ alignas(16) float bufI[3][kW];  // triple-buffered image rows (async-filled)
  __shared__ alignas(16) float bufD[2][kW];  // double-buffered depth rows
  __shared__ float s_time[kTS];
  __shared__ float s_occt[kT27];
  __shared__ int   s_kmap[kTS];   // nearest occ_t slice for each event time

  const size_t plane = (size_t)kH * kW;
  const size_t baseV = (size_t)b * kTS  * plane + (size_t)y * kW;  // + t*plane
  const size_t baseF = (size_t)b * kT27 * plane + (size_t)y * kW;  // + k*plane

  // Kick off the first async row copy immediately: overlaps with all scalar setup.
  feasai_async_cp(voxelgrid + baseV + x, &bufI[0][x]);             // prime t=0

  if (x < kTS)  s_time[x] = time[b * kTS + x];
  if (x < kT27) s_occt[x] = occ_t[b * kT27 + x];
  __syncthreads();

  if (x < kTS) {
    const float tt = s_time[x];
    int kk = 0; float best = fabsf(s_occt[0] - tt);
    for (int k = 1; k < kT27; ++k) {
      float d = fabsf(s_occt[k] - tt);
      if (d < best) { best = d; kk = k; }   // strict '<' => first-min (matches argmin)
    }
    s_kmap[x] = kk;
  }
  __syncthreads();

  const float reft = gt_t[b];
  const float fx00 = fx[b * 9];          // fx[:,0,0]
  const float vv   = v[b];
  const float sgn  = vv >= 0.0f ? 1.0f : -1.0f;  // v/|v|
  const float num  = fx00 * vv;                  // depth = num / (sgn * flow)

  // argmin |time - gt_t| and |occ_t - gt_t| (redundant per thread; trivial)
  int ev_idx = 0;
  { float best = fabsf(s_time[0] - reft);
    for (int t = 1; t < kTS; ++t) {
      float d = fabsf(s_time[t] - reft);
      if (d < best) { best = d; ev_idx = t; } } }
  int img_idx = 0;
  { float best = fabsf(s_occt[0] - reft);
    for (int k = 1; k < kT27; ++k) {
      float d = fabsf(s_occt[k] - reft);
      if (d < best) { best = d; img_idx = k; } } }

  // ---------------- event stack: 64 warped voxelgrid slices ----------------
  // Pipeline: read bufI[t%3], issue into bufI[(t+1)%3].  The buffer being
  // overwritten at iteration t was last READ at iteration t-2, and the single
  // barrier at iteration t-1 already separates those -> 1 barrier/iteration.
  float acc_ev = 0.0f;
  float ch3    = 0.0f;   // depth_64[b, ev_idx]

  for (int t = 0; t < kTS; ++t) {
    if (t + 1 < kTS)
      feasai_async_cp(voxelgrid + baseV + (size_t)(t + 1) * plane + x,
                      &bufI[(t + 1) % 3][x]);
    const float tt = s_time[t];
    const int   kk = s_kmap[t];
    const float f  = flow_27[baseF + (size_t)kk * plane + x] + EPSF;  // flow_64 + EPS
    const float shift = f * (tt - reft);                              // shift_ref_ev
    if (t + 1 < kTS) feasai_wait_async_le1(); else feasai_wait_async_le0();
    __syncthreads();                               // row t visible; prior reads done
    acc_ev += feasai_bilin(bufI[t % 3], (float)x - shift);
    if (t == ev_idx) ch3 = num / (sgn * f);
  }
  __syncthreads();                                 // ev reads done before img reuse

  // -------------- image stack: 27 warped occ_aps + depth_27 slices ---------
  float acc_img = 0.0f, acc_dep = 0.0f;
  float ch4 = 0.0f;   // depth_27[b, img_idx]

  feasai_async_cp(occ_aps + baseF + x, &bufI[0][x]);     // prime k=0
  for (int k = 0; k < kT27; ++k) {
    if (k + 1 < kT27)
      feasai_async_cp(occ_aps + baseF + (size_t)(k + 1) * plane + x,
                      &bufI[(k + 1) % 3][x]);
    const float ot = s_occt[k];
    const float f  = flow_27[baseF + (size_t)k * plane + x] + EPSF;   // flow_27p
    const float d  = num / (sgn * f);                                 // depth_27
    bufD[k & 1][x] = d;   // overwrite of bufD[k&1] at k+2 is fenced by barrier k+1
    const float shift = f * (ot - reft);                              // shift_ref_img
    if (k + 1 < kT27) feasai_wait_async_le1(); else feasai_wait_async_le0();
    __syncthreads();                               // occ row k + depth row visible
    const float xp = (float)x - shift;
    acc_img += feasai_bilin(bufI[k % 3], xp);
    acc_dep += feasai_bilin(bufD[k & 1], xp);
    if (k == img_idx) ch4 = d;
  }

  const float ch5 = depth_gt[baseF + (size_t)img_idx * plane + x];

  // -------------------------- outputs [bs,6,H,W] ---------------------------
  const size_t ob = ((size_t)b * 6) * plane + (size_t)y * kW + x;
  out[ob + 0 * plane] = acc_ev  / (float)kTS;
  out[ob + 1 * plane] = acc_img / (float)kT27;
  out[ob + 2 * plane] = acc_dep / (float)kT27;
  out[ob + 3 * plane] = ch3;
  out[ob + 4 * plane] = ch4;
  out[ob + 5 * plane] = ch5;
}

extern "C" void kernel_launch(void* const* d_in, const int* in_sizes, int n_in,
                              void* d_out, int out_size, void* d_ws, size_t ws_size,
                              hipStream_t stream) {
  const float* voxelgrid = (const float*)d_in[0];
  const float* time      = (const float*)d_in[1];
  const float* occ_aps   = (const float*)d_in[2];
  const float* occ_t     = (const float*)d_in[3];
  const float* gt_t      = (const float*)d_in[4];
  const float* fx        = (const float*)d_in[5];
  const float* v         = (const float*)d_in[6];
  const float* depth_gt  = (const float*)d_in[7];
  const float* flow_27   = (const float*)d_in[8];
  float* out = (float*)d_out;

  const int bs = in_sizes[4];            // gt_t has one entry per batch
  dim3 grid(kH, bs);                     // one block per (row y, batch b)
  dim3 block(kW);                        // one thread per pixel (8 wave32s)
  feasai_refocus_kernel<<<grid, block, 0, stream>>>(
      voxelgrid, time, occ_aps, occ_t, gt_t, fx, v, depth_gt, flow_27, out);
}